// ConnectedWithinCutoff_72224170049741
// MI455X (gfx1250) — compile-verified
//
#include <hip/hip_runtime.h>

// CDNA5 / gfx1250, wave32. Pairwise-distance + cutoff mask kernel.
//
// dist^2(i,j) = |pi|^2 - 2 pi.pj + |pj|^2 folded into ONE v_wmma_f32_16x16x4_f32
// per 16x16 tile via homogeneous coordinates:
//   A row m = (-2xi, -2yi, -2zi, 1)   (16x4, K=3 padded with the "1")
//   B col n = ( xj,   yj,   zj, |pj|^2)
//   C[m][n] = |pi|^2
// Output is 268 MB (> 192 MB L2, zero reuse) -> non-temporal stores; the whole
// kernel is store-bandwidth bound (~11.5 us at 23.3 TB/s), compute is noise.

typedef float v2f __attribute__((ext_vector_type(2)));
typedef float v8f __attribute__((ext_vector_type(8)));

#define CUTOFF2 25.0f  // cutoff = 5.0

__global__ __launch_bounds__(256) void cwc_zero_counts(int* ws, int B) {
    int t = blockIdx.x * blockDim.x + threadIdx.x;
    if (t < B) ws[t] = 0;
}

__global__ __launch_bounds__(256) void cwc_main(
    const float* __restrict__ pos,   // [B*n, 3]
    float*       __restrict__ out_pairs,  // [B*n*n, 2] as float
    float*       __restrict__ out_mask,   // [B*n*n]
    float*       __restrict__ out_dist,   // [B*n*n]
    int*         __restrict__ ws_counts,  // [B]
    int B, int n)
{
    const int tilesPerRow = n >> 4;              // n/16 column tiles
    const int b    = blockIdx.x / tilesPerRow;
    const int it   = blockIdx.x % tilesPerRow;   // 16-row strip of dist matrix
    const int i0   = it << 4;

    const int lane = threadIdx.x & 31;
    const int wave = threadIdx.x >> 5;           // 0..7
    const int lo   = lane & 15;
    const int hi   = lane >> 4;                  // 0 or 1

    const float* __restrict__ P = pos + (size_t)b * n * 3;

    // ---- A operand (constant across column tiles) ----
    // 32-bit A 16x4 layout: lanes 0-15 row m=lo hold K=0,1; lanes 16-31 hold K=2,3.
    {
        // nothing
    }
    const float aix = P[(i0 + lo) * 3 + 0];
    const float aiy = P[(i0 + lo) * 3 + 1];
    const float aiz = P[(i0 + lo) * 3 + 2];
    v2f aOp;
    aOp.x = (hi == 0) ? (-2.0f * aix) : (-2.0f * aiz);
    aOp.y = (hi == 0) ? (-2.0f * aiy) : 1.0f;     // K=3 homogeneous "1"

    // ---- C operand: c[v] = |p_{i0 + v + 8*hi}|^2 (f32 C/D layout) ----
    v8f cOp;
#pragma unroll
    for (int v = 0; v < 8; ++v) {
        const int m = v + 8 * hi;
        const float x = P[(i0 + m) * 3 + 0];
        const float y = P[(i0 + m) * 3 + 1];
        const float z = P[(i0 + m) * 3 + 2];
        cOp[v] = x * x + y * y + z * z;
    }

    int lane_count = 0;

    // Each of the 8 waves sweeps tilesPerRow/8 column tiles (uniform loop, EXEC all-1s).
    for (int jt = wave; jt < tilesPerRow; jt += 8) {
        const int j0 = jt << 4;

        // ---- B operand: lanes 0-15 col n=lo hold K=0,1; lanes 16-31 hold K=2,3 ----
        const float bjx = P[(j0 + lo) * 3 + 0];
        const float bjy = P[(j0 + lo) * 3 + 1];
        const float bjz = P[(j0 + lo) * 3 + 2];
        v2f bOp;
        bOp.x = (hi == 0) ? bjx : bjz;
        bOp.y = (hi == 0) ? bjy : (bjx * bjx + bjy * bjy + bjz * bjz); // K=3: |pj|^2

        // D = A*B + C = dist^2 for the 16x16 tile, one matrix op.
        v8f d = __builtin_amdgcn_wmma_f32_16x16x4_f32(
            /*neg_a=*/false, aOp, /*neg_b=*/false, bOp,
            /*c_mod=*/(short)0, cOp, /*reuse_a=*/false, /*reuse_b=*/false);

#pragma unroll
        for (int v = 0; v < 8; ++v) {
            const int i = i0 + v + 8 * hi;       // row of this D element
            const int j = j0 + lo;               // col of this D element
            float d2 = d[v];
            d2 = fmaxf(d2, 0.0f);                // guard tiny negatives from cancellation
            const float dist = __builtin_sqrtf(d2);
            const bool keep = (d2 <= CUTOFF2) && (i != j);

            const size_t idx = ((size_t)b * n + i) * (size_t)n + j;
            // 16 consecutive lanes write 16 consecutive floats -> coalesced 64B/128B.
            __builtin_nontemporal_store(dist, out_dist + idx);
            __builtin_nontemporal_store(keep ? 1.0f : 0.0f, out_mask + idx);
            v2f pr;
            pr.x = (float)(b * n + i);           // src (global node index)
            pr.y = (float)(b * n + j);           // dst
            __builtin_nontemporal_store(pr, (v2f*)out_pairs + idx);

            lane_count += keep ? 1 : 0;
        }
    }

    // Wave-level reduction of kept-edge count, single atomic per wave.
#pragma unroll
    for (int off = 16; off > 0; off >>= 1)
        lane_count += __shfl_xor(lane_count, off, 32);
    if (lane == 0) atomicAdd(&ws_counts[b], lane_count);
}

__global__ __launch_bounds__(256) void cwc_finalize_counts(
    const int* __restrict__ ws, float* __restrict__ out_nedges, int B) {
    int t = blockIdx.x * blockDim.x + threadIdx.x;
    if (t < B) out_nedges[t] = (float)ws[t];
}

extern "C" void kernel_launch(void* const* d_in, const int* in_sizes, int n_in,
                              void* d_out, int out_size, void* d_ws, size_t ws_size,
                              hipStream_t stream) {
    // Inputs per setup_inputs(): d_in[0] = num_nodes [B] int32 (unused: equal-size
    // graphs, n derived from shapes), d_in[1] = positions [B*n,3] float32.
    const int B = in_sizes[0];
    const int n = in_sizes[1] / (3 * B);   // 512

    const float* pos = (const float*)d_in[1];
    float* out = (float*)d_out;

    const size_t total = (size_t)B * n * n;
    float* out_pairs  = out;                      // [2*total] edge_index_all
    float* out_mask   = out + 2 * total;          // [total]   edge_mask
    float* out_nedges = out + 3 * total;          // [B]       num_edges
    float* out_dist   = out + 3 * total + B;      // [total]   distances

    int* ws_counts = (int*)d_ws;                  // B ints of scratch

    cwc_zero_counts<<<(B + 255) / 256, 256, 0, stream>>>(ws_counts, B);

    const int tilesPerRow = n / 16;               // 32
    cwc_main<<<B * tilesPerRow, 256, 0, stream>>>(
        pos, out_pairs, out_mask, out_dist, ws_counts, B, n);

    cwc_finalize_counts<<<(B + 255) / 256, 256, 0, stream>>>(ws_counts, out_nedges, B);
}